// Model_85925115724062
// MI455X (gfx1250) — compile-verified
//
#include <hip/hip_runtime.h>
#include <hip/hip_bf16.h>

// LightGCN-style 3-layer GCN propagation + L2 normalize, tuned for MI455X:
//  - SpMM lives in the 192MB L2 (x + x_next = 153.6MB): hw fp32 atomics, NT edge streams
//  - global_prefetch_b8 of next gather row
//  - final row-norms via V_WMMA_F32_16X16X4_F32 (16-row tile Gram diagonal, exact fp32)

#define USER_N 100000
#define ITEM_N 50000
#define ATTR_N 50000
#define N_TOT  150000
#define DIM    128
#define NLAYER 3
#define EPW    8   // edges per wave in SpMM

typedef float v2f __attribute__((ext_vector_type(2)));
typedef float v8f __attribute__((ext_vector_type(8)));

__global__ void zero4_kernel(float4* __restrict__ p, long n4) {
  long i = (long)blockIdx.x * blockDim.x + threadIdx.x;
  if (i < n4) p[i] = make_float4(0.f, 0.f, 0.f, 0.f);
}

__global__ void deg_kernel(const int* __restrict__ rows, const float* __restrict__ vals,
                           float* __restrict__ deg, int nE) {
  int e = blockIdx.x * blockDim.x + threadIdx.x;
  if (e < nE) unsafeAtomicAdd(&deg[rows[e]], __builtin_nontemporal_load(vals + e));
}

__global__ void dinv_kernel(const float* __restrict__ deg, float* __restrict__ dinv, int n) {
  int i = blockIdx.x * blockDim.x + threadIdx.x;
  if (i < n) { float d = deg[i]; dinv[i] = d > 0.f ? rsqrtf(d) : 0.f; }
}

__global__ void normv_kernel(const int* __restrict__ rows, const int* __restrict__ cols,
                             const float* __restrict__ vals, const float* __restrict__ dinv,
                             float* __restrict__ nv, int nE) {
  int e = blockIdx.x * blockDim.x + threadIdx.x;
  if (e < nE) {
    int r = __builtin_nontemporal_load(rows + e);
    int c = __builtin_nontemporal_load(cols + e);
    float v = __builtin_nontemporal_load(vals + e);
    __builtin_nontemporal_store(v * dinv[r] * dinv[c], nv + e);
  }
}

// x = concat(users, items): rows [0,USER) <- embed[ITEM+ATTR+row], rows [USER,N) <- embed[row-USER]
__global__ void build_x_kernel(const float* __restrict__ emb, float* __restrict__ x) {
  long t = (long)blockIdx.x * blockDim.x + threadIdx.x;      // one float4 per thread
  if (t >= (long)N_TOT * (DIM / 4)) return;
  long row = t >> 5; int q = (int)(t & 31);
  long src = (row < USER_N) ? (ITEM_N + ATTR_N + row) : (row - USER_N);
  ((float4*)(x + row * DIM))[q] = ((const float4*)(emb + src * DIM))[q];
}

// One wave per EPW consecutive edges; lane L owns float4 slice [4L,4L+4) of the 128-wide row.
__global__ void spmm_kernel(const int* __restrict__ rows, const int* __restrict__ cols,
                            const float* __restrict__ nv, const float* __restrict__ x,
                            float* __restrict__ xn, int nE) {
  const int lane = threadIdx.x & 31;
  const long wid = (long)blockIdx.x * (blockDim.x >> 5) + (threadIdx.x >> 5);
  long base = wid * EPW;
  if (base >= nE) return;
  int c_next = __builtin_nontemporal_load(cols + base);
  for (int k = 0; k < EPW; ++k) {
    long e = base + k;
    if (e >= nE) break;                                   // uniform across wave
    const int   c = c_next;
    const int   r = __builtin_nontemporal_load(rows + e);
    const float w = __builtin_nontemporal_load(nv + e);
    if (k + 1 < EPW && e + 1 < nE) {
      c_next = __builtin_nontemporal_load(cols + e + 1);
      __builtin_prefetch(x + (size_t)c_next * DIM + lane * 4, 0, 3); // global_prefetch_b8
    }
    const float4 v = ((const float4*)(x + (size_t)c * DIM))[lane];
    float* dst = xn + (size_t)r * DIM + lane * 4;
    unsafeAtomicAdd(dst + 0, w * v.x);                    // global_atomic_add_f32, no-return
    unsafeAtomicAdd(dst + 1, w * v.y);
    unsafeAtomicAdd(dst + 2, w * v.z);
    unsafeAtomicAdd(dst + 3, w * v.w);
  }
}

__global__ void acc_add_kernel(float4* __restrict__ out, const float4* __restrict__ xn, long n4) {
  long i = (long)blockIdx.x * blockDim.x + threadIdx.x;
  if (i < n4) {
    float4 a = out[i]; float4 b = xn[i];
    a.x += b.x; a.y += b.y; a.z += b.z; a.w += b.w;
    out[i] = a;
  }
}

// Per 16-row tile: Gram(16x16) of emb=out/3 via 32x V_WMMA_F32_16X16X4_F32,
// diagonal -> squared norms -> scale rows in place.
__global__ void finalize_kernel(float* __restrict__ out, int nrows) {
  const int lane = threadIdx.x & 31;
  const int wib  = threadIdx.x >> 5;
  const int tile = blockIdx.x * (blockDim.x >> 5) + wib;
  __shared__ float s_scale[4][16];

  const bool valid = ((long)tile * 16 < nrows);
  const float inv3 = 1.0f / (float)NLAYER;
  v8f acc = {};
  int R = tile * 16;
  const int r = lane & 15;      // row within tile (both A and B operands)
  const int h = lane >> 4;      // lane-half selects K pairs {2h, 2h+1} per VGPR

  if (valid) {
    // lane holds E[R+r][kb + 2h + v], v in {0,1} -> same data is A (16x4) and B=A^T (4x16)
    const float2* p2 = (const float2*)(out + (size_t)(R + r) * DIM + 2 * h);
#pragma unroll
    for (int i = 0; i < DIM / 4; ++i) {
      float2 f = p2[2 * i];
      v2f a; a.x = f.x * inv3; a.y = f.y * inv3;
      acc = __builtin_amdgcn_wmma_f32_16x16x4_f32(false, a, false, a,
                                                  (short)0, acc, false, false);
    }
    // Diagonal D[m][m]: m=0..7 at (vgpr m, lane m); m=8..15 at (vgpr m-8, lane m+16)
    int q = (lane < 8) ? lane : ((lane >= 24) ? (lane - 24) : 0);
    float d = acc[0];
    d = (q == 1) ? acc[1] : d;  d = (q == 2) ? acc[2] : d;
    d = (q == 3) ? acc[3] : d;  d = (q == 4) ? acc[4] : d;
    d = (q == 5) ? acc[5] : d;  d = (q == 6) ? acc[6] : d;
    d = (q == 7) ? acc[7] : d;
    if (lane < 8 || lane >= 24) {
      int m = (lane < 8) ? lane : (lane - 16);
      float nrm = sqrtf(fmaxf(d, 0.0f));
      s_scale[wib][m] = 1.0f / fmaxf(nrm, 1e-12f);
    }
  }
  __syncthreads();
  if (valid) {
#pragma unroll
    for (int m = 0; m < 16; ++m) {
      float sc = s_scale[wib][m] * inv3;
      float4* p = (float4*)(out + (size_t)(R + m) * DIM) + lane;
      float4 v = *p;
      v.x *= sc; v.y *= sc; v.z *= sc; v.w *= sc;
      *p = v;
    }
  }
}

extern "C" void kernel_launch(void* const* d_in, const int* in_sizes, int n_in,
                              void* d_out, int out_size, void* d_ws, size_t ws_size,
                              hipStream_t stream) {
  const float* emb  = (const float*)d_in[0];
  const int*   rows = (const int*)d_in[1];
  const int*   cols = (const int*)d_in[2];
  const float* vals = (const float*)d_in[3];
  const int    nE   = in_sizes[1];

  float* out = (float*)d_out;
  float* ws  = (float*)d_ws;
  // workspace layout (floats): deg[N] | dinv[N] | nv[E] | xA[N*128] | xB[N*128]
  size_t offE = ((size_t)nE + 3) & ~(size_t)3;
  float* deg  = ws;
  float* dinv = ws + N_TOT;
  float* nv   = ws + 2 * (size_t)N_TOT;
  float* xA   = ws + 2 * (size_t)N_TOT + offE;
  float* xB   = xA + (size_t)N_TOT * DIM;

  const int  T = 256;
  const long n4x = (long)N_TOT * (DIM / 4);            // float4 count of a feature matrix

  // degrees + normalized edge weights
  zero4_kernel<<<(N_TOT / 4 + T - 1) / T, T, 0, stream>>>((float4*)deg, N_TOT / 4);
  deg_kernel<<<(nE + T - 1) / T, T, 0, stream>>>(rows, vals, deg, nE);
  dinv_kernel<<<(N_TOT + T - 1) / T, T, 0, stream>>>(deg, dinv, N_TOT);
  normv_kernel<<<(nE + T - 1) / T, T, 0, stream>>>(rows, cols, vals, dinv, nv, nE);

  // x0 = concat(users, items); acc (= d_out) starts at zero
  build_x_kernel<<<(int)((n4x + T - 1) / T), T, 0, stream>>>(emb, xA);
  zero4_kernel<<<(int)((n4x + T - 1) / T), T, 0, stream>>>((float4*)out, n4x);

  // 3 propagation layers (num_gcn_layer fixed at 3; graph-capture forbids reading device scalar)
  float* xs = xA; float* xd = xB;
  const int spmmBlocks = (nE + (EPW * (T / 32)) - 1) / (EPW * (T / 32));
  for (int l = 0; l < NLAYER; ++l) {
    zero4_kernel<<<(int)((n4x + T - 1) / T), T, 0, stream>>>((float4*)xd, n4x);
    spmm_kernel<<<spmmBlocks, T, 0, stream>>>(rows, cols, nv, xs, xd, nE);
    acc_add_kernel<<<(int)((n4x + T - 1) / T), T, 0, stream>>>((float4*)out, (const float4*)xd, n4x);
    float* t = xs; xs = xd; xd = t;
  }

  // emb = acc/3, row-L2-normalize in place (WMMA Gram diagonal)
  const int tiles  = N_TOT / 16;                       // 9375
  const int fThr   = 128;                              // 4 waves -> 4 tiles per block
  const int fBlks  = (tiles + (fThr / 32) - 1) / (fThr / 32);
  finalize_kernel<<<fBlks, fThr, 0, stream>>>(out, N_TOT);
}